// HSA_35948876267692
// MI455X (gfx1250) — compile-verified
//
#include <hip/hip_runtime.h>

// ---------------------------------------------------------------------------
// Fused dynamic-filter kernel for MI455X (gfx1250, wave32).
//   stage 1: filters = relu(conv3x3_{48->9}(pixel_unshuffle(i_t)) + b)   [WMMA fp32]
//   stage 2: out[b,c,p] = sum_f patch3x3(h)[b,c,f,p] * filters[b,f,p]    [VALU]
// One block = one (batch, row, 128-pixel) segment; filters kept in LDS.
// ---------------------------------------------------------------------------

typedef __attribute__((ext_vector_type(2))) float v2f;
typedef __attribute__((ext_vector_type(8))) float v8f;

#define NB      2      // batch
#define NF      64     // h channels
#define HH      256
#define WW      256
#define KTOT    432    // 48 input channels * 9 taps
#define WROW    434    // padded LDS row stride for weights (even, bank-friendly)
#define TILEX   128    // pixels per block (one row segment)

__global__ __launch_bounds__(256)
void HSA_dynfilter_fused(const float* __restrict__ hsrc,
                         const float* __restrict__ i_t,
                         const float* __restrict__ conv_w,
                         const float* __restrict__ conv_b,
                         float* __restrict__ out)
{
    __shared__ float wlds[16 * WROW];      // reordered weights W[o][k], k = tap*48 + i
    __shared__ float flds[TILEX][13];      // per-pixel 9 filter taps (padded stride)

    const int tid  = threadIdx.x;
    const int wave = tid >> 5;
    const int lane = tid & 31;
    const int l15  = lane & 15;            // A: out-channel row / B,D: pixel column
    const int hi   = lane >> 4;            // 0 -> K pair (0,1), 1 -> K pair (2,3)

    // block -> (b, y, xseg)
    const int bx    = blockIdx.x;
    const int xseg  = bx & 1;              // WW / TILEX = 2
    const int y     = (bx >> 1) & (HH - 1);
    const int b     = bx >> 9;
    const int xbase = xseg * TILEX;

    const float* hbat = hsrc + (long)b * NF * HH * WW;

    // --- prefetch the h tile this block will stream in stage 2 (overlap with WMMA)
    for (int idx = tid; idx < NF * 3 * 8; idx += 256) {
        int cc  = idx / 24;
        int rem = idx - cc * 24;
        int dyi = rem >> 3;                // 0..2
        int seg = rem & 7;                 // 8 x 16 floats = 128-pixel row
        int yy  = y + dyi - 1;
        if ((unsigned)yy < (unsigned)HH)
            __builtin_prefetch(hbat + (long)cc * HH * WW + yy * WW + xbase + seg * 16, 0, 1);
    }

    // --- stage weights into LDS, reordered k = tap*48 + i, zero-padded to 16 rows
    for (int idx = tid; idx < 16 * KTOT; idx += 256) {
        int o = idx / KTOT;
        int k = idx - o * KTOT;
        float v = 0.f;
        if (o < 9) {
            int tap = k / 48;              // kh*3 + kw
            int i   = k - tap * 48;        // shuffled input channel
            v = conv_w[(o * 48 + i) * 9 + tap];
        }
        wlds[o * WROW + k] = v;
    }
    __syncthreads();

    // ------------------------------------------------------------------
    // Stage 1: each wave computes 16 pixels x 16 out-channels via
    // 108 x V_WMMA_F32_16X16X4_F32 over K = 432 (= 9 taps x 12 c-groups).
    // Per lane, per K-step: ONE global_load_b64 of exactly the K-pair this
    // lane feeds (lanes 0-15: fx 0,1; lanes 16-31: fx 2,3) from a clamped
    // (always in-bounds) address, zeroed by 2 cndmasks when out-of-range.
    // Weights: ds_load_b64 at constant offsets. Two alternating accumulators.
    // ------------------------------------------------------------------
    const int x = xbase + wave * 16 + l15;            // this lane's pixel column
    const float* itb  = i_t + (long)b * 3 * 1024 * 1024;
    const float* wrow = &wlds[l15 * WROW + (hi ? 2 : 0)];

    v8f acc0 = {}, acc1 = {};
#pragma unroll
    for (int tap = 0; tap < 9; ++tap) {
        const int dy = tap / 3 - 1;                   // compile-time per unrolled tap
        const int dx = tap % 3 - 1;
        const int yy = y + dy;                        // wave-uniform
        const int xx = x + dx;                        // per-lane
        const int yyc = yy < 0 ? 0 : (yy > HH - 1 ? HH - 1 : yy);   // SALU clamp
        const int xxc = xx < 0 ? 0 : (xx > WW - 1 ? WW - 1 : xx);   // v_med3
        const bool ok = ((unsigned)yy < (unsigned)HH) & ((unsigned)xx < (unsigned)WW);

        // clamped base: element (b, c=0, yyc*4, xxc*4 + 2*hi) of i_t
        const float* p = itb + (long)(yyc * 4) * 1024 + xxc * 4 + 2 * hi;

        v2f qq[12];
#pragma unroll
        for (int c = 0; c < 3; ++c)
#pragma unroll
            for (int fy = 0; fy < 4; ++fy) {
                v2f t = *(const v2f*)(p + (long)c * 1048576 + fy * 1024);
                qq[c * 4 + fy][0] = ok ? t[0] : 0.f;
                qq[c * 4 + fy][1] = ok ? t[1] : 0.f;
            }

#pragma unroll
        for (int j = 0; j < 12; ++j) {
            v2f afrag = *(const v2f*)(wrow + tap * 48 + 4 * j);
            if (j & 1)
                acc1 = __builtin_amdgcn_wmma_f32_16x16x4_f32(
                    false, afrag, false, qq[j], (short)0, acc1, false, false);
            else
                acc0 = __builtin_amdgcn_wmma_f32_16x16x4_f32(
                    false, afrag, false, qq[j], (short)0, acc0, false, false);
        }
    }
    v8f acc = acc0 + acc1;

    // D layout: VGPR r -> (o=r, px=lane) for lanes 0-15, (o=r+8, px=lane-16) for 16-31
    {
        const int pxb = wave * 16 + l15;
        if (hi == 0) {
#pragma unroll
            for (int r = 0; r < 8; ++r) {
                float v = acc[r] + conv_b[r];
                flds[pxb][r] = v > 0.f ? v : 0.f;
            }
        } else {
            float v = acc[0] + conv_b[8];             // o = 8; o>=9 are padding
            flds[pxb][8] = v > 0.f ? v : 0.f;
        }
    }
    __syncthreads();

    // ------------------------------------------------------------------
    // Stage 2: dynamic per-pixel 3x3 conv over all 64 channels of h.
    // wave-strided (channel, 32-pixel chunk) iteration, coalesced loads/stores.
    // ------------------------------------------------------------------
    for (int it = wave; it < NF * 4; it += 8) {
        const int cc    = it >> 2;
        const int chunk = it & 3;
        const int xl    = chunk * 32 + lane;          // 0..127
        const int xg    = xbase + xl;
        const float* hp = hbat + (long)cc * HH * WW;

        float s = 0.f;
#pragma unroll
        for (int dy = -1; dy <= 1; ++dy) {
            const int yy = y + dy;
            const bool yok = (unsigned)yy < (unsigned)HH;
#pragma unroll
            for (int dx = -1; dx <= 1; ++dx) {
                const int xx = xg + dx;
                float hv = 0.f;
                if (yok && (unsigned)xx < (unsigned)WW) hv = hp[yy * WW + xx];
                s += hv * flds[xl][(dy + 1) * 3 + (dx + 1)];
            }
        }
        out[(long)b * NF * HH * WW + (long)cc * HH * WW + y * WW + xg] = s;
    }
}

extern "C" void kernel_launch(void* const* d_in, const int* in_sizes, int n_in,
                              void* d_out, int out_size, void* d_ws, size_t ws_size,
                              hipStream_t stream) {
    (void)in_sizes; (void)n_in; (void)out_size; (void)d_ws; (void)ws_size;
    const float* h  = (const float*)d_in[0];   // (2, 64, 256, 256)
    const float* it = (const float*)d_in[1];   // (2, 3, 1024, 1024)
    const float* cw = (const float*)d_in[2];   // (9, 48, 3, 3)
    const float* cb = (const float*)d_in[3];   // (9,)
    float* out = (float*)d_out;                // (2, 64, 256, 256)

    const int blocks = NB * HH * (WW / TILEX); // 1024
    HSA_dynfilter_fused<<<blocks, 256, 0, stream>>>(h, it, cw, cb, out);
}